// LinearAttention_15685220565712
// MI455X (gfx1250) — compile-verified
//
#include <hip/hip_runtime.h>
#include <hip/hip_bf16.h>
#include <math.h>

// ---------------------------------------------------------------------------
// Shapes (fixed by the reference):
//   x: (16, 256, 64, 64) f32, w_qkv: (384, 256) f32, w_out: (256, 128) f32
//   B=16, C=256, N=64*64=4096, HEADS=4, DIM_HEAD=32, HIDDEN=128, O3=384
//
// Pipeline (all matrix ops = V_WMMA_F32_16X16X4_F32, wave32):
//   1. qkv = Wqkv @ x                    (16x [384x256]@[256x4096])
//   2. q   = softmax_n(q)    in-place    (register-resident, 1R+1W)
//   3. k   = softmax_d(q)                (coalesced column softmax)
//   4. ctx = k @ v^T                     (K-split 8 ways for wave count)
//   5. W2[b] = Wout (x)_h ctx[b,h]^T     (fold ctx into output weights, 2MiB)
//   6. out = W2[b] @ q                   (16x [256x128]@[128x4096])
// Step 5+6 eliminate the 64 MiB attn round-trip entirely.
// ---------------------------------------------------------------------------
#define BATCH   16
#define CIN     256
#define NPIX    4096
#define HEADS   4
#define DHEAD   32
#define HIDDEN  128
#define O3      384
#define KSPLIT  8                 // K-split for the context GEMM
#define KCHUNK  (NPIX / KSPLIT)   // 512

typedef __attribute__((ext_vector_type(2))) float v2f;
typedef __attribute__((ext_vector_type(8))) float v8f;

// D = A(16x4 f32) * B(4x16 f32) + C(16x16 f32), wave32 WMMA.
__device__ __forceinline__ v8f wmma4(v2f a, v2f b, v8f c) {
  return __builtin_amdgcn_wmma_f32_16x16x4_f32(false, a, false, b, (short)0, c,
                                               false, false);
}

// ---------------------------------------------------------------------------
// Kernel 1: QKV projection.  qkv[b,o,n] = sum_c wqkv[o,c] * x[b,c,n]
// 8 waves; each wave: one 16-wide n-tile, THREE 16-row m-tiles
// {w*16, w*16+128, w*16+256} -> block covers all 384 output rows per n-tile,
// so x streams from HBM exactly once; wqkv (384 KiB) stays L2-resident.
// ---------------------------------------------------------------------------
__global__ __launch_bounds__(256) void k_qkv_gemm(
    const float* __restrict__ x, const float* __restrict__ wqkv,
    float* __restrict__ qkv) {
  const int lane = threadIdx.x & 31;
  const int wave = threadIdx.x >> 5;
  const int half = lane >> 4;   // 0/1
  const int lm   = lane & 15;   // 0..15
  const int nBase = blockIdx.x * 16;
  const int b     = blockIdx.y;

  const float* __restrict__ xb = x + (size_t)b * CIN * NPIX;
  float* __restrict__ qb = qkv + (size_t)b * O3 * NPIX;

  const int m0 = wave * 16;
  v8f acc0 = {}, acc1 = {}, acc2 = {};

  for (int kb = 0; kb < CIN; kb += 4) {
    const int k0 = kb + 2 * half;
    v2f bf, a0, a1, a2;
    bf.x = xb[(size_t)k0 * NPIX + nBase + lm];
    bf.y = xb[(size_t)(k0 + 1) * NPIX + nBase + lm];
    a0.x = wqkv[(m0 + lm) * CIN + k0];
    a0.y = wqkv[(m0 + lm) * CIN + k0 + 1];
    a1.x = wqkv[(m0 + 128 + lm) * CIN + k0];
    a1.y = wqkv[(m0 + 128 + lm) * CIN + k0 + 1];
    a2.x = wqkv[(m0 + 256 + lm) * CIN + k0];
    a2.y = wqkv[(m0 + 256 + lm) * CIN + k0 + 1];
    acc0 = wmma4(a0, bf, acc0);
    acc1 = wmma4(a1, bf, acc1);
    acc2 = wmma4(a2, bf, acc2);
  }
#pragma unroll
  for (int v = 0; v < 8; ++v) {
    const int row = v + 8 * half;
    qb[(size_t)(m0 + row) * NPIX + nBase + lm]       = acc0[v];
    qb[(size_t)(m0 + 128 + row) * NPIX + nBase + lm] = acc1[v];
    qb[(size_t)(m0 + 256 + row) * NPIX + nBase + lm] = acc2[v];
  }
}

// ---------------------------------------------------------------------------
// Kernel 2: softmax over n (axis=-1) of q, in place.  One block per row;
// the 4096-float row lives in registers (16 per thread) -> exactly one
// global read + one global write per element.
// ---------------------------------------------------------------------------
__global__ __launch_bounds__(256) void k_softmax_n(float* __restrict__ qkv) {
  const int r = blockIdx.x;          // 0 .. 16*128-1
  const int b = r >> 7;
  const int c = r & 127;
  float* __restrict__ row = qkv + ((size_t)b * O3 + c) * NPIX;

  __shared__ float red[256];
  const int t = threadIdx.x;

  float v[16];
  float m = -INFINITY;
#pragma unroll
  for (int j = 0; j < 16; ++j) {
    v[j] = row[t + j * 256];
    m = fmaxf(m, v[j]);
  }
  red[t] = m; __syncthreads();
  for (int s = 128; s > 0; s >>= 1) {
    if (t < s) red[t] = fmaxf(red[t], red[t + s]);
    __syncthreads();
  }
  m = red[0]; __syncthreads();

  float sum = 0.f;
#pragma unroll
  for (int j = 0; j < 16; ++j) {
    v[j] = __expf(v[j] - m);
    sum += v[j];
  }
  red[t] = sum; __syncthreads();
  for (int s = 128; s > 0; s >>= 1) {
    if (t < s) red[t] += red[t + s];
    __syncthreads();
  }
  const float inv = 1.0f / red[0];
#pragma unroll
  for (int j = 0; j < 16; ++j) row[t + j * 256] = v[j] * inv;
}

// ---------------------------------------------------------------------------
// Kernel 3: k = softmax over d (axis=-2) of softmaxed q.  One thread per
// (b,h,n) column of 32 strided elements; lanes walk consecutive n -> all
// loads/stores fully coalesced.
// ---------------------------------------------------------------------------
__global__ __launch_bounds__(256) void k_softmax_d(
    const float* __restrict__ qkv, float* __restrict__ kbuf) {
  const int col = blockIdx.x * 256 + threadIdx.x;  // 0 .. 16*4*4096-1
  const int n = col & (NPIX - 1);
  const int h = (col >> 12) & (HEADS - 1);
  const int b = col >> 14;

  const float* __restrict__ qs = qkv + ((size_t)b * O3 + h * DHEAD) * NPIX + n;
  float vals[DHEAD];
  float m = -INFINITY;
#pragma unroll
  for (int d = 0; d < DHEAD; ++d) {
    vals[d] = qs[(size_t)d * NPIX];
    m = fmaxf(m, vals[d]);
  }
  float s = 0.f;
#pragma unroll
  for (int d = 0; d < DHEAD; ++d) {
    vals[d] = __expf(vals[d] - m);
    s += vals[d];
  }
  const float inv = 1.0f / s;
  float* __restrict__ ko =
      kbuf + (size_t)(b * HEADS + h) * DHEAD * NPIX + n;
#pragma unroll
  for (int d = 0; d < DHEAD; ++d) ko[(size_t)d * NPIX] = vals[d] * inv;
}

// ---------------------------------------------------------------------------
// Kernel 4a: partial context.  pctx[ks, bh, kd, vd] =
//   sum_{n in chunk ks} k[bh,kd,n] * v[bh,vd,n]
// K=4096 split KSPLIT ways -> 8*64 blocks * 4 waves = 2048 waves in flight
// to cover latency on the 64 MiB of HBM reads.
// ---------------------------------------------------------------------------
__global__ __launch_bounds__(128) void k_context_part(
    const float* __restrict__ kbuf, const float* __restrict__ qkv,
    float* __restrict__ pctx) {
  const int ks = blockIdx.x;   // 0..KSPLIT-1
  const int bh = blockIdx.y;   // 0..63
  const int b = bh >> 2, h = bh & 3;
  const int lane = threadIdx.x & 31;
  const int wave = threadIdx.x >> 5;
  const int half = lane >> 4, lm = lane & 15;
  const int mBase = (wave >> 1) * 16;  // kd tile
  const int nBase = (wave & 1) * 16;   // vd tile
  const int kBeg  = ks * KCHUNK;

  const float* __restrict__ kA = kbuf + (size_t)bh * DHEAD * NPIX;  // [kd][n]
  const float* __restrict__ vB =
      qkv + ((size_t)b * O3 + 2 * HIDDEN + h * DHEAD) * NPIX;       // [vd][n]

  v8f acc = {};
  for (int kb = 0; kb < KCHUNK; kb += 4) {
    const int k0 = kBeg + kb + 2 * half;
    v2f a, bb;
    a.x  = kA[(size_t)(mBase + lm) * NPIX + k0];
    a.y  = kA[(size_t)(mBase + lm) * NPIX + k0 + 1];
    bb.x = vB[(size_t)(nBase + lm) * NPIX + k0];  // B[k,col] = v[col,k]
    bb.y = vB[(size_t)(nBase + lm) * NPIX + k0 + 1];
    acc = wmma4(a, bb, acc);
  }
#pragma unroll
  for (int v = 0; v < 8; ++v) {
    const int row = mBase + v + 8 * half;
    pctx[(((size_t)ks * (BATCH * HEADS) + bh) * DHEAD + row) * DHEAD + nBase +
         lm] = acc[v];
  }
}

// ---------------------------------------------------------------------------
// Kernel 4b: reduce the KSPLIT partial contexts.  65536 elements total.
// ---------------------------------------------------------------------------
__global__ __launch_bounds__(256) void k_ctx_reduce(
    const float* __restrict__ pctx, float* __restrict__ ctx) {
  const int i = blockIdx.x * 256 + threadIdx.x;  // 0 .. 65535
  const size_t stride = (size_t)BATCH * HEADS * DHEAD * DHEAD;
  float s = 0.f;
#pragma unroll
  for (int ks = 0; ks < KSPLIT; ++ks) s += pctx[(size_t)ks * stride + i];
  ctx[i] = s;
}

// ---------------------------------------------------------------------------
// Kernel 5: fold context into the output weights.
//   W2[b][o, h*32+kd] = sum_vd wout[o, h*32+vd] * ctx[b,h][kd, vd]
// One block (8 waves) per bh.  Wave w: m-tiles {w*16, w*16+128}, both
// 16-wide kd tiles -> 4 accumulators, K=32 over vd.
// ---------------------------------------------------------------------------
__global__ __launch_bounds__(256) void k_fold_w2(
    const float* __restrict__ wout, const float* __restrict__ ctx,
    float* __restrict__ w2) {
  const int bh = blockIdx.x;
  const int b = bh >> 2, h = bh & 3;
  const int lane = threadIdx.x & 31;
  const int wave = threadIdx.x >> 5;
  const int half = lane >> 4, lm = lane & 15;
  const int m0 = wave * 16;

  const float* __restrict__ cb = ctx + (size_t)bh * DHEAD * DHEAD;  // [kd][vd]

  v8f acc00 = {}, acc01 = {}, acc10 = {}, acc11 = {};
#pragma unroll
  for (int kb = 0; kb < DHEAD; kb += 4) {
    const int k0 = kb + 2 * half;    // vd
    v2f a0, a1, b0, b1;
    a0.x = wout[(m0 + lm) * HIDDEN + h * DHEAD + k0];
    a0.y = wout[(m0 + lm) * HIDDEN + h * DHEAD + k0 + 1];
    a1.x = wout[(m0 + 128 + lm) * HIDDEN + h * DHEAD + k0];
    a1.y = wout[(m0 + 128 + lm) * HIDDEN + h * DHEAD + k0 + 1];
    // B[vd, kd] = ctx[kd][vd]
    b0.x = cb[(size_t)(0 + lm) * DHEAD + k0];
    b0.y = cb[(size_t)(0 + lm) * DHEAD + k0 + 1];
    b1.x = cb[(size_t)(16 + lm) * DHEAD + k0];
    b1.y = cb[(size_t)(16 + lm) * DHEAD + k0 + 1];
    acc00 = wmma4(a0, b0, acc00);
    acc01 = wmma4(a0, b1, acc01);
    acc10 = wmma4(a1, b0, acc10);
    acc11 = wmma4(a1, b1, acc11);
  }
  float* __restrict__ w2b = w2 + (size_t)b * CIN * HIDDEN;
#pragma unroll
  for (int v = 0; v < 8; ++v) {
    const int row = v + 8 * half;
    w2b[(size_t)(m0 + row) * HIDDEN + h * DHEAD + 0 + lm]        = acc00[v];
    w2b[(size_t)(m0 + row) * HIDDEN + h * DHEAD + 16 + lm]       = acc01[v];
    w2b[(size_t)(m0 + 128 + row) * HIDDEN + h * DHEAD + 0 + lm]  = acc10[v];
    w2b[(size_t)(m0 + 128 + row) * HIDDEN + h * DHEAD + 16 + lm] = acc11[v];
  }
}

// ---------------------------------------------------------------------------
// Kernel 6: fused output GEMM.  out[b,o,n] = sum_c W2[b][o,c] * q[b,c,n]
// q is read straight from the (softmaxed) qkv buffer -- no attn round trip.
// 8 waves x 2 m-tiles = all 256 output rows per block, q streamed once.
// ---------------------------------------------------------------------------
__global__ __launch_bounds__(256) void k_out_gemm(
    const float* __restrict__ qkv, const float* __restrict__ w2,
    float* __restrict__ out) {
  const int lane = threadIdx.x & 31;
  const int wave = threadIdx.x >> 5;
  const int half = lane >> 4, lm = lane & 15;
  const int nBase = blockIdx.x * 16;
  const int b     = blockIdx.y;

  const float* __restrict__ qs  = qkv + (size_t)b * O3 * NPIX;   // q rows 0..127
  const float* __restrict__ w2b = w2 + (size_t)b * CIN * HIDDEN;
  float* __restrict__ ob = out + (size_t)b * CIN * NPIX;

  const int m0 = wave * 16;
  v8f acc0 = {}, acc1 = {};
  for (int kb = 0; kb < HIDDEN; kb += 4) {
    const int k0 = kb + 2 * half;
    v2f bf, a0, a1;
    bf.x = qs[(size_t)k0 * NPIX + nBase + lm];
    bf.y = qs[(size_t)(k0 + 1) * NPIX + nBase + lm];
    a0.x = w2b[(m0 + lm) * HIDDEN + k0];
    a0.y = w2b[(m0 + lm) * HIDDEN + k0 + 1];
    a1.x = w2b[(m0 + 128 + lm) * HIDDEN + k0];
    a1.y = w2b[(m0 + 128 + lm) * HIDDEN + k0 + 1];
    acc0 = wmma4(a0, bf, acc0);
    acc1 = wmma4(a1, bf, acc1);
  }
#pragma unroll
  for (int v = 0; v < 8; ++v) {
    const int row = v + 8 * half;
    ob[(size_t)(m0 + row) * NPIX + nBase + lm]       = acc0[v];
    ob[(size_t)(m0 + 128 + row) * NPIX + nBase + lm] = acc1[v];
  }
}

// ---------------------------------------------------------------------------
extern "C" void kernel_launch(void* const* d_in, const int* in_sizes, int n_in,
                              void* d_out, int out_size, void* d_ws,
                              size_t ws_size, hipStream_t stream) {
  const float* x    = (const float*)d_in[0];  // (16,256,64,64)
  const float* wqkv = (const float*)d_in[1];  // (384,256)
  const float* wout = (const float*)d_in[2];  // (256,128)
  float* out = (float*)d_out;                 // (16,256,64,64)

  // Workspace layout (floats):
  //   qkv : 16*384*4096 = 25,165,824   (96 MiB)
  //   kbuf: 16*128*4096 =  8,388,608   (32 MiB)
  //   ctx : 16*4*32*32  =     65,536   (256 KiB)
  //   pctx: 8*65,536    =    524,288   (  2 MiB)
  //   w2  : 16*256*128  =    524,288   (  2 MiB)
  char* ws = (char*)d_ws;
  const size_t QKV_F = (size_t)BATCH * O3 * NPIX;
  const size_t KB_F  = (size_t)BATCH * HIDDEN * NPIX;
  const size_t CTX_F = (size_t)BATCH * HEADS * DHEAD * DHEAD;
  float* qkv  = (float*)ws;
  float* kbuf = (float*)(ws + QKV_F * 4);
  float* ctx  = (float*)(ws + (QKV_F + KB_F) * 4);
  float* pctx = (float*)(ws + (QKV_F + KB_F + CTX_F) * 4);
  float* w2   = (float*)(ws + (QKV_F + KB_F + CTX_F + KSPLIT * CTX_F) * 4);

  k_qkv_gemm<<<dim3(NPIX / 16, BATCH), 256, 0, stream>>>(x, wqkv, qkv);
  k_softmax_n<<<BATCH * HIDDEN, 256, 0, stream>>>(qkv);
  k_softmax_d<<<(BATCH * HEADS * NPIX) / 256, 256, 0, stream>>>(qkv, kbuf);
  k_context_part<<<dim3(KSPLIT, BATCH * HEADS), 128, 0, stream>>>(kbuf, qkv,
                                                                  pctx);
  k_ctx_reduce<<<CTX_F / 256, 256, 0, stream>>>(pctx, ctx);
  k_fold_w2<<<BATCH * HEADS, 256, 0, stream>>>(wout, ctx, w2);
  k_out_gemm<<<dim3(NPIX / 16, BATCH), 256, 0, stream>>>(qkv, w2, out);
}